// TripletLoss_88519275970910
// MI455X (gfx1250) — compile-verified
//
#include <hip/hip_runtime.h>
#include <math.h>

#define MARGIN 0.2f
#define DIM 512

typedef __attribute__((ext_vector_type(2))) float v2f;
typedef __attribute__((ext_vector_type(8))) float v8f;
typedef __attribute__((__vector_size__(4 * sizeof(int)))) int v4i;

// ---------------------------------------------------------------------------
// gfx1250 async global->LDS copy (ASYNCcnt path), with safe fallback.
// Builtin signature (from hipcc diagnostic): param0 = v4i in AS(1) (global),
// param1 = v4i in AS(3) (LDS), then i32 offset, i32 cpol.
// ---------------------------------------------------------------------------
#if defined(__has_builtin)
#if __has_builtin(__builtin_amdgcn_global_load_async_to_lds_b128)
#define TL_ASYNC 1
#endif
#endif

__device__ __forceinline__ void tl_copy_b128(const float* gsrc, float* ldst) {
#ifdef TL_ASYNC
    __builtin_amdgcn_global_load_async_to_lds_b128(
        (__attribute__((address_space(1))) v4i*)gsrc,
        (__attribute__((address_space(3))) v4i*)ldst,
        /*offset=*/0, /*cpol=*/0);
#else
    *(float4*)ldst = *(const float4*)gsrc;
#endif
}

__device__ __forceinline__ void tl_wait_async() {
#ifdef TL_ASYNC
#if __has_builtin(__builtin_amdgcn_s_wait_asynccnt)
    __builtin_amdgcn_s_wait_asynccnt(0);
#else
    asm volatile("s_wait_asynccnt 0x0" ::: "memory");
#endif
#endif
}

// ---------------------------------------------------------------------------
// Workspace layout (4-byte units):
//   [0] float total   [1] uint count
//   [4..4+N) n1   [4+N..) n2   [4+2N..) pos_sq   [4+3N..) d_pos
// ---------------------------------------------------------------------------

__global__ void tl_init(float* ws) {
    ws[0] = 0.0f;
    ((unsigned int*)ws)[1] = 0u;
}

// One block (128 threads) per row: each thread loads one float4 of each embed.
__global__ void tl_row_stats(const float* __restrict__ e1,
                             const float* __restrict__ e2,
                             float* __restrict__ n1, float* __restrict__ n2,
                             float* __restrict__ ps, float* __restrict__ dp) {
    const int row  = blockIdx.x;
    const int tid  = threadIdx.x;          // 0..127
    const int lane = tid & 31;
    const int wave = tid >> 5;             // 0..3

    const float4 a = ((const float4*)(e1 + (size_t)row * DIM))[tid];
    const float4 b = ((const float4*)(e2 + (size_t)row * DIM))[tid];

    float s1 = a.x*a.x + a.y*a.y + a.z*a.z + a.w*a.w;
    float s2 = b.x*b.x + b.y*b.y + b.z*b.z + b.w*b.w;
    float dx = a.x-b.x, dy = a.y-b.y, dz = a.z-b.z, dw = a.w-b.w;
    float sp = dx*dx + dy*dy + dz*dz + dw*dw;

    #pragma unroll
    for (int off = 16; off > 0; off >>= 1) {
        s1 += __shfl_down(s1, off, 32);
        s2 += __shfl_down(s2, off, 32);
        sp += __shfl_down(sp, off, 32);
    }

    __shared__ float red[3][4];
    if (lane == 0) { red[0][wave] = s1; red[1][wave] = s2; red[2][wave] = sp; }
    __syncthreads();
    if (tid == 0) {
        float t1 = red[0][0] + red[0][1] + red[0][2] + red[0][3];
        float t2 = red[1][0] + red[1][1] + red[1][2] + red[1][3];
        float tp = red[2][0] + red[2][1] + red[2][2] + red[2][3];
        n1[row] = t1;
        n2[row] = t2;
        ps[row] = tp;
        dp[row] = sqrtf(tp);
    }
}

// ---------------------------------------------------------------------------
// Fused Gram + triplet hinge kernel, double-buffered LDS with async staging.
//   blockIdx.z: z=0 D11(e1,e1) z=1 D12(e1,e2) z=2 D21(e2,e1) z=3 D22(e2,e2)
//   Block tile 128x64; 8 waves (4x2); per-wave 32x32 = four 16x16 fp32 accs.
// ---------------------------------------------------------------------------
#define BT_M 128
#define BT_N 64
#define KT   32
#define KST  36   // LDS row stride (floats): 144B -> 16B aligned, conflict-free

__global__ __launch_bounds__(256)
void tl_gram_loss(const float* __restrict__ e1,
                  const float* __restrict__ e2,
                  const float* __restrict__ n1, const float* __restrict__ n2,
                  const float* __restrict__ ps, const float* __restrict__ dp,
                  float* __restrict__ totalAcc, unsigned int* __restrict__ cntAcc,
                  int N) {
    __shared__ float sA[2][BT_M * KST];
    __shared__ float sB[2][BT_N * KST];

    const int z = blockIdx.z;
    const float* __restrict__ A  = (z & 2) ? e2 : e1;   // row operand
    const float* __restrict__ B  = (z & 1) ? e2 : e1;   // col operand
    const float* __restrict__ nA = (z & 2) ? n2 : n1;
    const float* __restrict__ nB = (z & 1) ? n2 : n1;

    const int rowBase = blockIdx.y * BT_M;
    const int colBase = blockIdx.x * BT_N;

    const int tid     = threadIdx.x;        // 0..255
    const int lane    = tid & 31;
    const int wave    = tid >> 5;           // 0..7
    const int waveRow = wave >> 1;          // 0..3
    const int waveCol = wave & 1;           // 0..1
    const int l16     = lane & 15;
    const int half    = lane >> 4;          // 0 or 1

    // Precomputed staging indices (one b128 lane-copy each).
    int ar[4], ac[4], br[2], bc[2];
    #pragma unroll
    for (int i = 0; i < 4; ++i) { int f = tid + i * 256; ar[i] = f >> 3; ac[i] = (f & 7) * 4; }
    #pragma unroll
    for (int i = 0; i < 2; ++i) { int f = tid + i * 256; br[i] = f >> 3; bc[i] = (f & 7) * 4; }

    // Fragment base offsets (ISA 16x4 fp32 A/B layout: lanes 0-15 read
    // K={k,k+1}, lanes 16-31 read K={k+2,k+3} of row l16).
    const int aOff0 = (waveRow * 32 + l16) * KST + 2 * half;
    const int aOff1 = aOff0 + 16 * KST;
    const int bOff0 = (waveCol * 32 + l16) * KST + 2 * half;
    const int bOff1 = bOff0 + 16 * KST;

    v8f acc[2][2] = {};

    // Prologue: stage tile kb=0 into buffer 0.
    #pragma unroll
    for (int i = 0; i < 4; ++i)
        tl_copy_b128(A + (size_t)(rowBase + ar[i]) * DIM + ac[i], &sA[0][ar[i] * KST + ac[i]]);
    #pragma unroll
    for (int i = 0; i < 2; ++i)
        tl_copy_b128(B + (size_t)(colBase + br[i]) * DIM + bc[i], &sB[0][br[i] * KST + bc[i]]);
    tl_wait_async();
    __syncthreads();

    int cur = 0;
    for (int kb = 0; kb < DIM; kb += KT) {
        const int nxt = cur ^ 1;
        // Async prefetch of next K-tile overlaps the WMMA burst below.
        if (kb + KT < DIM) {
            #pragma unroll
            for (int i = 0; i < 4; ++i)
                tl_copy_b128(A + (size_t)(rowBase + ar[i]) * DIM + (kb + KT) + ac[i],
                             &sA[nxt][ar[i] * KST + ac[i]]);
            #pragma unroll
            for (int i = 0; i < 2; ++i)
                tl_copy_b128(B + (size_t)(colBase + br[i]) * DIM + (kb + KT) + bc[i],
                             &sB[nxt][br[i] * KST + bc[i]]);
        }

        const float* aBase0 = &sA[cur][aOff0];
        const float* aBase1 = &sA[cur][aOff1];
        const float* bBase0 = &sB[cur][bOff0];
        const float* bBase1 = &sB[cur][bOff1];

        #pragma unroll
        for (int k = 0; k < KT; k += 4) {
            v2f a0 = *(const v2f*)(aBase0 + k);
            v2f a1 = *(const v2f*)(aBase1 + k);
            v2f b0 = *(const v2f*)(bBase0 + k);
            v2f b1 = *(const v2f*)(bBase1 + k);
            acc[0][0] = __builtin_amdgcn_wmma_f32_16x16x4_f32(
                false, a0, false, b0, (short)0, acc[0][0], false, false);
            acc[0][1] = __builtin_amdgcn_wmma_f32_16x16x4_f32(
                false, a0, false, b1, (short)0, acc[0][1], false, false);
            acc[1][0] = __builtin_amdgcn_wmma_f32_16x16x4_f32(
                false, a1, false, b0, (short)0, acc[1][0], false, false);
            acc[1][1] = __builtin_amdgcn_wmma_f32_16x16x4_f32(
                false, a1, false, b1, (short)0, acc[1][1], false, false);
        }

        tl_wait_async();     // prefetch landed in LDS
        __syncthreads();     // all waves done reading `cur`, staging visible
        cur = nxt;
    }

    // Fused epilogue: dist -> mask -> hinge, accumulated per lane.
    float        total = 0.0f;
    unsigned int cnt   = 0u;

    const int wgRow = rowBase + waveRow * 32;
    const int wgCol = colBase + waveCol * 32;

    #pragma unroll
    for (int rt = 0; rt < 2; ++rt) {
        #pragma unroll
        for (int ct = 0; ct < 2; ++ct) {
            #pragma unroll
            for (int r = 0; r < 8; ++r) {
                // C/D layout: VGPR r -> M = r (lanes 0-15) / 8+r (lanes 16-31)
                int gi = wgRow + rt * 16 + r + 8 * half;
                int gj = wgCol + ct * 16 + l16;
                float g    = acc[rt][ct][r];
                float dist = nA[gi] + nB[gj] - 2.0f * g;
                dist = fmaxf(dist, 0.0f);
                if (gi != gj) {
                    float p = ps[gi];
                    if (p < dist) {
                        float hin = MARGIN + dp[gi] - sqrtf(dist);
                        total += fmaxf(hin, 0.0f);
                        cnt   += 1u;
                    }
                }
            }
        }
    }

    #pragma unroll
    for (int off = 16; off > 0; off >>= 1) {
        total += __shfl_down(total, off, 32);
        cnt   += __shfl_down(cnt, off, 32);
    }
    if (lane == 0) {
        atomicAdd(totalAcc, total);
        atomicAdd(cntAcc, cnt);
    }
}

__global__ void tl_finalize(const float* ws, float* out) {
    float total = ws[0];
    float count = (float)((const unsigned int*)ws)[1];
    out[0] = total / fmaxf(count, 1.0f);
}

extern "C" void kernel_launch(void* const* d_in, const int* in_sizes, int n_in,
                              void* d_out, int out_size, void* d_ws, size_t ws_size,
                              hipStream_t stream) {
    const float* e1 = (const float*)d_in[0];
    const float* e2 = (const float*)d_in[1];
    const int N = in_sizes[0] / DIM;   // 4096

    float* ws    = (float*)d_ws;
    float* total = ws;                                   // [0]
    unsigned int* cnt = (unsigned int*)d_ws + 1;         // [1]
    float* n1 = ws + 4;
    float* n2 = n1 + N;
    float* ps = n2 + N;
    float* dp = ps + N;

    tl_init<<<1, 1, 0, stream>>>(ws);
    tl_row_stats<<<N, 128, 0, stream>>>(e1, e2, n1, n2, ps, dp);

    dim3 grid(N / BT_N, N / BT_M, 4);
    tl_gram_loss<<<grid, 256, 0, stream>>>(e1, e2, n1, n2, ps, dp,
                                           total, cnt, N);

    tl_finalize<<<1, 1, 0, stream>>>(ws, (float*)d_out);
}